// ZeroPadder_19490561590035
// MI455X (gfx1250) — compile-verified
//
#include <hip/hip_runtime.h>

// out[b,t,:] = [ x[b,t,:], zeros(1024) ]  for x:[16,2048,1024] f32.
// weight = [eye|0] so the matmul is bitwise a zero-pad copy.
// Streaming: 128 MiB read + 256 MiB write = 384 MiB -> ~16.5 us at 23.3 TB/s.
// Copy half: CDNA5 async global->LDS->global B128 path, 8 ops in flight per
// wave before the ASYNCcnt wait (MLP for the 23.3 TB/s * ~600ns BW-delay
// product). Zero half: NT B128 stores. NT everywhere: zero reuse, >L2 footprint.

typedef float v4f __attribute__((ext_vector_type(4)));

#define THREADS     256u
#define ROWS        (16u * 2048u)     // 32768 output rows
#define V4_ROW_OUT  512u              // 2048 f32 per out row
#define V4_ROW_X    256u              // 1024 f32 per x row
#define RPB         8u                // rows per block (8-deep async pipeline)
#define COPY_BLOCKS (ROWS / RPB)      // 4096
#define GRID        (2u * COPY_BLOCKS)

__global__ __launch_bounds__(THREADS) void zeropad_kernel(
    const float* __restrict__ x, float* __restrict__ out) {
  __shared__ v4f stage[RPB * THREADS];   // 32 KB: one B128 slot per lane per row
  const unsigned t = threadIdx.x;
  const unsigned b = blockIdx.x;
  v4f* __restrict__ out4 = (v4f*)out;

  if (b < COPY_BLOCKS) {
    // ---- copy: rows [8b, 8b+8) of x -> out[:, 0:1024] ----
    const v4f* __restrict__ x4 = (const v4f*)x;
    const unsigned base = b * (RPB * V4_ROW_X);   // f4 index into x
    const unsigned row0 = b * RPB;

#pragma unroll
    for (unsigned k = 0; k < RPB; ++k) {
      unsigned lds = (unsigned)(size_t)&stage[k * THREADS + t];
      const v4f* src = x4 + base + k * V4_ROW_X + t;
      asm volatile("global_load_async_to_lds_b128 %0, %1, off th:TH_LOAD_NT"
                   :: "v"(lds), "v"((unsigned long long)(size_t)src)
                   : "memory");
    }
#if __has_builtin(__builtin_amdgcn_s_wait_asynccnt)
    __builtin_amdgcn_s_wait_asynccnt(0);   // all 8 LDS slots populated
#else
    asm volatile("s_wait_asynccnt 0" ::: "memory");
#endif
#pragma unroll
    for (unsigned k = 0; k < RPB; ++k) {
      unsigned lds = (unsigned)(size_t)&stage[k * THREADS + t];
      v4f* dst = out4 + (size_t)(row0 + k) * V4_ROW_OUT + t;
      asm volatile("global_store_async_from_lds_b128 %0, %1, off th:TH_STORE_NT"
                   :: "v"((unsigned long long)(size_t)dst), "v"(lds)
                   : "memory");
    }
    // outstanding ASYNCcnt drained by implicit wait-idle at s_endpgm
  } else {
    // ---- zero: rows [8b', 8b'+8) of out[:, 1024:2048] ----
    const unsigned row0 = (b - COPY_BLOCKS) * RPB;
    v4f z = {0.f, 0.f, 0.f, 0.f};
#pragma unroll
    for (unsigned k = 0; k < RPB; ++k) {
      __builtin_nontemporal_store(
          z, out4 + (size_t)(row0 + k) * V4_ROW_OUT + V4_ROW_X + t);
    }
  }
}

extern "C" void kernel_launch(void* const* d_in, const int* in_sizes, int n_in,
                              void* d_out, int out_size, void* d_ws, size_t ws_size,
                              hipStream_t stream) {
  const float* x = (const float*)d_in[0];
  // d_in[1] (weight) is identity||zeros: matmul == copy + zero-pad, bitwise
  // identical in f32, so we never read the 8 MB weight.
  float* out = (float*)d_out;
  (void)in_sizes; (void)n_in; (void)out_size; (void)d_ws; (void)ws_size;

  zeropad_kernel<<<dim3(GRID), dim3(THREADS), 0, stream>>>(x, out);
}